// SAMInsOpt_58961311040345
// MI455X (gfx1250) — compile-verified
//
#include <hip/hip_runtime.h>
#include <math.h>

// ---------------------------------------------------------------------------
// SAMInsOpt: 50 Adam iterations of deformable-registration loss on gfx1250.
//   param [1,3,32,32,32], feat_source/target [1,128,32,32,32] (all f32).
// Per iteration:
//   disp = 9x separable 3-tap zero-pad mean of param        (pool9 kernel)
//   fwd+bwd of cosine data term through trilinear sampling   (data_kernel,
//       channel reductions done with V_WMMA_F32_16X16X4_F32)
//   + smoothness-reg gradient                                (regcombine)
//   grad_param = pool9(grad_disp)   (operator is self-adjoint)
//   Adam update.
// All L2-resident (32MB feats << 192MB L2); f32 throughout for fidelity.
// ---------------------------------------------------------------------------

#define NVOX  32768                 // 32^3
#define NP    (3 * NVOX)            // param elements
#define NCH   128

typedef __attribute__((ext_vector_type(2))) float v2f;
typedef __attribute__((ext_vector_type(8))) float v8f;

// ---------------------------------------------------------------------------
// pool9: nine chained 1-D 3-tap zero-padded means (axes z,y,x repeated 3x),
// entirely in LDS (2 x 128KB ping-pong buffers of the 320KB WGP LDS).
// One workgroup per channel (grid = 3), 1024 threads, 32 elems/thread.
// Self-adjoint => reused for the backward pass.
// ---------------------------------------------------------------------------
__global__ __launch_bounds__(1024) void pool9_kernel(const float* __restrict__ in,
                                                     float* __restrict__ out) {
  extern __shared__ float lds[];           // 2 * 32768 floats
  float* A = lds;
  float* B = lds + NVOX;
  const int ch  = blockIdx.x;
  const int tid = threadIdx.x;

  for (int i = tid; i < NVOX; i += 1024) A[i] = in[ch * NVOX + i];
  __syncthreads();

  const int strides[3] = {1024, 32, 1};    // z, y, x
  for (int p = 0; p < 9; ++p) {
    const int s = strides[p % 3];
    for (int i = tid; i < NVOX; i += 1024) {
      const int pos = (i / s) & 31;
      const float l = (pos > 0)  ? A[i - s] : 0.0f;
      const float r = (pos < 31) ? A[i + s] : 0.0f;
      B[i] = (l + A[i] + r) * (1.0f / 3.0f);
    }
    __syncthreads();
    float* t = A; A = B; B = t;
  }
  for (int i = tid; i < NVOX; i += 1024) out[ch * NVOX + i] = A[i];
}

// ---------------------------------------------------------------------------
// data_kernel: fused forward + backward of the data term.
// One wave32 per 16 voxels. lane = (half<<4)|m : voxel m, channel parity half.
// Phase A: trilinear-sample 128 channels into LDS tiles W[16][132], T[16][132]
//          (pitch 132 -> LDS bank stride 4, conflict free), plus scalar
//          partials Sum|w| and Sum t^2.
// Phase B: Swt = diag(W T^T), Sww = diag(W W^T) via V_WMMA_F32_16X16X4_F32,
//          K accumulated over 32 chunks of 4. A frag: lane=row, VGPRs walk K,
//          lanes 16-31 carry K+2. B frag: lane=column, mirrored -> both frags
//          load with the identical LDS indexing.
// Phase C: dL/dw_c = cT*t_c + cW*w_c  (L1-normalize projection term is
//          exactly 0 since g is orthogonal to wn); re-gather corners, form
//          trilinear coordinate derivatives, reduce to grad wrt disp.
//          raw = z + 15.5*disp -> chain factor 15.5 * clip-mask.
// ---------------------------------------------------------------------------
__global__ __launch_bounds__(32) void data_kernel(const float* __restrict__ fsrc,
                                                  const float* __restrict__ ftgt,
                                                  const float* __restrict__ disp,
                                                  float* __restrict__ gdata) {
  __shared__ float sW[16][132];
  __shared__ float sT[16][132];
  __shared__ float scr[8][32];

  const int lane = threadIdx.x;
  const int m    = lane & 15;     // voxel within tile
  const int half = lane >> 4;     // channel parity
  const int v    = blockIdx.x * 16 + m;
  const int z = v >> 10, y = (v >> 5) & 31, x = v & 31;

  // phi = grid0 + disp ; unnorm: raw = idx + 15.5*disp ; clip to [0,31]
  const float rz = (float)z + 15.5f * disp[v];
  const float ry = (float)y + 15.5f * disp[NVOX + v];
  const float rx = (float)x + 15.5f * disp[2 * NVOX + v];
  const float mz = (rz >= 0.0f && rz <= 31.0f) ? 1.0f : 0.0f;
  const float my = (ry >= 0.0f && ry <= 31.0f) ? 1.0f : 0.0f;
  const float mx = (rx >= 0.0f && rx <= 31.0f) ? 1.0f : 0.0f;
  const float cz = fminf(fmaxf(rz, 0.0f), 31.0f);
  const float cy = fminf(fmaxf(ry, 0.0f), 31.0f);
  const float cx = fminf(fmaxf(rx, 0.0f), 31.0f);
  const float fz = cz - floorf(cz), fy = cy - floorf(cy), fx = cx - floorf(cx);
  const int z0 = (int)floorf(cz), y0 = (int)floorf(cy), x0 = (int)floorf(cx);
  const int z1 = min(z0 + 1, 31), y1 = min(y0 + 1, 31), x1 = min(x0 + 1, 31);

  const int o000 = (z0 * 32 + y0) * 32 + x0, o001 = (z0 * 32 + y0) * 32 + x1;
  const int o010 = (z0 * 32 + y1) * 32 + x0, o011 = (z0 * 32 + y1) * 32 + x1;
  const int o100 = (z1 * 32 + y0) * 32 + x0, o101 = (z1 * 32 + y0) * 32 + x1;
  const int o110 = (z1 * 32 + y1) * 32 + x0, o111 = (z1 * 32 + y1) * 32 + x1;

  // ---- Phase A: sample, fill tiles, scalar partial reductions --------------
  float s1p = 0.0f, sttp = 0.0f;
  for (int it = 0; it < 64; ++it) {
    const int c = 2 * it + half;
    const float* f = fsrc + c * NVOX;
    const float f000 = f[o000], f001 = f[o001], f010 = f[o010], f011 = f[o011];
    const float f100 = f[o100], f101 = f[o101], f110 = f[o110], f111 = f[o111];
    const float c00 = f000 + (f001 - f000) * fx;
    const float c01 = f010 + (f011 - f010) * fx;
    const float c10 = f100 + (f101 - f100) * fx;
    const float c11 = f110 + (f111 - f110) * fx;
    const float c0 = c00 + (c01 - c00) * fy;
    const float c1 = c10 + (c11 - c10) * fy;
    const float w  = c0 + (c1 - c0) * fz;
    const float t  = ftgt[c * NVOX + v];
    sW[m][c] = w;
    sT[m][c] = t;
    s1p  += fabsf(w);
    sttp += t * t;
  }
  __syncthreads();
  const float S1  = s1p  + __shfl_xor(s1p, 16, 32);
  const float Stt = sttp + __shfl_xor(sttp, 16, 32);

  // ---- Phase B: WMMA cross-block reductions, extract diagonals -------------
  v8f accWT = {0.f, 0.f, 0.f, 0.f, 0.f, 0.f, 0.f, 0.f};
  v8f accWW = {0.f, 0.f, 0.f, 0.f, 0.f, 0.f, 0.f, 0.f};
  const int kb = half * 2;
#pragma unroll 4
  for (int k = 0; k < 32; ++k) {
    const int c0 = 4 * k + kb;
    v2f a, b;
    a.x = sW[m][c0]; a.y = sW[m][c0 + 1];   // A: row m of W (K walks VGPRs)
    b.x = sT[m][c0]; b.y = sT[m][c0 + 1];   // B: column m of T^T (same idx)
    accWT = __builtin_amdgcn_wmma_f32_16x16x4_f32(false, a, false, b,
                                                  (short)0, accWT, false, false);
    accWW = __builtin_amdgcn_wmma_f32_16x16x4_f32(false, a, false, a,
                                                  (short)0, accWW, false, false);
  }
  // C/D layout: VGPR r holds M=r (lanes 0-15, N=lane) and M=r+8 (lanes 16-31).
#pragma unroll
  for (int r = 0; r < 8; ++r) scr[r][lane] = accWT[r];
  __syncthreads();
  const float Swt = (m < 8) ? scr[m][m] : scr[m - 8][m + 16];
  __syncthreads();
#pragma unroll
  for (int r = 0; r < 8; ++r) scr[r][lane] = accWW[r];
  __syncthreads();
  const float Sww = (m < 8) ? scr[m][m] : scr[m - 8][m + 16];

  // ---- Phase C: per-voxel scalars, re-gather, coordinate gradient ----------
  const float L1 = fmaxf(S1, 1e-12f);
  const float av = Swt / L1;                       // a = sum(wn*t)
  const float bb = Sww / (L1 * L1);                // sum(wn^2)
  const float bn = fmaxf(sqrtf(bb), 1e-8f);
  const float tn = fmaxf(sqrtf(Stt), 1e-8f);
  const float cT = -1.0f / (32768.0f * bn * tn * L1);
  const float cW = -cT * (av / (bn * bn)) / L1;

  float az = 0.0f, ay = 0.0f, ax = 0.0f;
  for (int it = 0; it < 64; ++it) {
    const int c = 2 * it + half;
    const float* f = fsrc + c * NVOX;
    const float f000 = f[o000], f001 = f[o001], f010 = f[o010], f011 = f[o011];
    const float f100 = f[o100], f101 = f[o101], f110 = f[o110], f111 = f[o111];
    const float g = cT * sT[m][c] + cW * sW[m][c];
    const float b00 = f000 + (f001 - f000) * fx;   // z0,y0
    const float b01 = f010 + (f011 - f010) * fx;   // z0,y1
    const float b10 = f100 + (f101 - f100) * fx;   // z1,y0
    const float b11 = f110 + (f111 - f110) * fx;   // z1,y1
    const float ddz = (b10 + (b11 - b10) * fy) - (b00 + (b01 - b00) * fy);
    const float ddy = (b01 - b00) * (1.0f - fz) + (b11 - b10) * fz;
    const float ddx = ((f001 - f000) * (1.0f - fy) + (f011 - f010) * fy) * (1.0f - fz)
                    + ((f101 - f100) * (1.0f - fy) + (f111 - f110) * fy) * fz;
    az += g * ddz; ay += g * ddy; ax += g * ddx;
  }
  az += __shfl_xor(az, 16, 32);
  ay += __shfl_xor(ay, 16, 32);
  ax += __shfl_xor(ax, 16, 32);
  if (half == 0) {
    gdata[v]             = az * 15.5f * mz;
    gdata[NVOX + v]      = ay * 15.5f * my;
    gdata[2 * NVOX + v]  = ax * 15.5f * mx;
  }
}

// ---------------------------------------------------------------------------
// regcombine: grad_total = grad_data + d(reg)/d(disp)
// reg = 1000 * sum_axes mean(diff^2), each diff tensor has 3*31*32*32 = 95232
// elements; d/d d[k] = 2*lambda/N * ((d[k]-d[k-1])[k>0] - (d[k+1]-d[k])[k<31]).
// ---------------------------------------------------------------------------
__global__ void regcombine_kernel(const float* __restrict__ disp,
                                  const float* __restrict__ gdata,
                                  float* __restrict__ gtotal) {
  const int i = blockIdx.x * blockDim.x + threadIdx.x;
  if (i >= NP) return;
  const int v = i & (NVOX - 1);
  const int z = v >> 10, y = (v >> 5) & 31, x = v & 31;
  const float d = disp[i];
  const float coef = 2.0f * 1000.0f / 95232.0f;
  float g = gdata[i];
  if (z > 0)  g += coef * (d - disp[i - 1024]);
  if (z < 31) g -= coef * (disp[i + 1024] - d);
  if (y > 0)  g += coef * (d - disp[i - 32]);
  if (y < 31) g -= coef * (disp[i + 32] - d);
  if (x > 0)  g += coef * (d - disp[i - 1]);
  if (x < 31) g -= coef * (disp[i + 1] - d);
  gtotal[i] = g;
}

// ---------------------------------------------------------------------------
// Adam update. Bias-correction factors precomputed on host per step.
// ---------------------------------------------------------------------------
__global__ void adam_kernel(float* __restrict__ p, float* __restrict__ m,
                            float* __restrict__ v, const float* __restrict__ g,
                            float bc1, float bc2) {
  const int i = blockIdx.x * blockDim.x + threadIdx.x;
  if (i >= NP) return;
  const float gi = g[i];
  const float mi = 0.9f * m[i] + 0.1f * gi;
  const float vi = 0.999f * v[i] + 0.001f * gi * gi;
  m[i] = mi;
  v[i] = vi;
  const float mh = mi / bc1;
  const float vh = vi / bc2;
  p[i] -= 0.05f * mh / (sqrtf(vh) + 1e-4f);
}

// ---------------------------------------------------------------------------
extern "C" void kernel_launch(void* const* d_in, const int* in_sizes, int n_in,
                              void* d_out, int out_size, void* d_ws, size_t ws_size,
                              hipStream_t stream) {
  const float* param_in = (const float*)d_in[0];
  const float* fsrc     = (const float*)d_in[1];
  const float* ftgt     = (const float*)d_in[2];
  float* out = (float*)d_out;

  float* ws     = (float*)d_ws;
  float* p      = ws;
  float* m      = ws + 1 * NP;
  float* v      = ws + 2 * NP;
  float* disp   = ws + 3 * NP;
  float* gdata  = ws + 4 * NP;
  float* gtotal = ws + 5 * NP;
  float* gparam = ws + 6 * NP;

  hipMemcpyAsync(p, param_in, NP * sizeof(float), hipMemcpyDeviceToDevice, stream);
  hipMemsetAsync(m, 0, NP * sizeof(float), stream);
  hipMemsetAsync(v, 0, NP * sizeof(float), stream);

  const size_t poolLds = 2 * NVOX * sizeof(float);   // 256 KB of 320 KB WGP LDS
  for (int t = 1; t <= 50; ++t) {
    pool9_kernel<<<3, 1024, poolLds, stream>>>(p, disp);
    data_kernel<<<NVOX / 16, 32, 0, stream>>>(fsrc, ftgt, disp, gdata);
    regcombine_kernel<<<(NP + 255) / 256, 256, 0, stream>>>(disp, gdata, gtotal);
    pool9_kernel<<<3, 1024, poolLds, stream>>>(gtotal, gparam);  // self-adjoint
    const float bc1 = 1.0f - powf(0.9f, (float)t);
    const float bc2 = 1.0f - powf(0.999f, (float)t);
    adam_kernel<<<(NP + 255) / 256, 256, 0, stream>>>(p, m, v, gparam, bc1, bc2);
  }

  hipMemcpyAsync(out, p, NP * sizeof(float), hipMemcpyDeviceToDevice, stream);
}